// BiMPMMatching_68204080661133
// MI455X (gfx1250) — compile-verified
//
#include <hip/hip_runtime.h>
#include <hip/hip_bf16.h>

#define B    32
#define LSEQ 128
#define H    200
#define NP   20
#define DOUT 105
#define EPS  1e-8f
#define NEGV -1e7f

typedef float v2f __attribute__((ext_vector_type(2)));
typedef float v8f __attribute__((ext_vector_type(8)));

__device__ __forceinline__ float wsum32(float v) {
#pragma unroll
    for (int o = 16; o > 0; o >>= 1) v += __shfl_xor(v, o, 32);
    return v;
}
// order-preserving float->uint key for atomicMax over signed floats
__device__ __forceinline__ unsigned fkey(float f) {
    unsigned u = __float_as_uint(f);
    return (u & 0x80000000u) ? ~u : (u | 0x80000000u);
}
__device__ __forceinline__ float funkey(unsigned k) {
    return __uint_as_float((k & 0x80000000u) ? (k & 0x7fffffffu) : ~k);
}

// ---------------- K0: masked context, norms, weighted norms -----------------
// grid = B*LSEQ, block = 32 (one wave)
__global__ void k_pre(const float* __restrict__ ctx, const float* __restrict__ mask,
                      const float* __restrict__ wmp,
                      float* __restrict__ cm, float* __restrict__ nrm,
                      float* __restrict__ wn) {
    int bl = blockIdx.x;
    int b = bl / LSEQ, l = bl % LSEQ;
    int lane = threadIdx.x;
    float mk = mask[bl];
    const float* src = ctx + (size_t)bl * H;
    float* dst = cm + (size_t)bl * H;
    float s = 0.f;
    float wa[NP];
#pragma unroll
    for (int k = 0; k < NP; k++) wa[k] = 0.f;
    for (int h = lane; h < H; h += 32) {
        float x = src[h] * mk;
        dst[h] = x;
        float x2 = x * x;
        s += x2;
#pragma unroll
        for (int k = 0; k < NP; k++) {
            float w = wmp[k * H + h];
            wa[k] += w * w * x2;
        }
    }
    s = wsum32(s);
#pragma unroll
    for (int k = 0; k < NP; k++) wa[k] = wsum32(wa[k]);
    if (lane == 0) {
        nrm[bl] = sqrtf(s);
#pragma unroll
        for (int k = 0; k < NP; k++)
            wn[((size_t)b * NP + k) * LSEQ + l] = sqrtf(wa[k]);
    }
}

// ---------------- K1: last tokens + mask sums -----------------
// grid = B, block = 256
__global__ void k_last(const float* __restrict__ mask_p, const float* __restrict__ mask_h,
                       const float* __restrict__ cpm, const float* __restrict__ chm,
                       float* __restrict__ last_p, float* __restrict__ last_h,
                       float* __restrict__ summ_p, float* __restrict__ summ_h) {
    int b = blockIdx.x;
    __shared__ int ips, ihs;
    if (threadIdx.x == 0) {
        float sp = 0.f, sh = 0.f;
        for (int i = 0; i < LSEQ; i++) { sp += mask_p[b * LSEQ + i]; sh += mask_h[b * LSEQ + i]; }
        summ_p[b] = sp; summ_h[b] = sh;
        int ip = (int)sp - 1; if (ip < 0) ip = 0;
        int ih = (int)sh - 1; if (ih < 0) ih = 0;
        ips = ip; ihs = ih;
    }
    __syncthreads();
    int ip = ips, ih = ihs;
    for (int h = threadIdx.x; h < H; h += blockDim.x) {
        last_p[b * H + h] = cpm[((size_t)b * LSEQ + ip) * H + h];
        last_h[b * H + h] = chm[((size_t)b * LSEQ + ih) * H + h];
    }
}

// ---------------- K2: cosine matrix via WMMA f32 16x16x4 -----------------
// grid = B*8 (row strip), block = 256 (8 waves -> 8 col tiles)
__global__ void k_cos(const float* __restrict__ cpm, const float* __restrict__ chm,
                      const float* __restrict__ np_, const float* __restrict__ nh_,
                      float* __restrict__ cosm) {
    int blk = blockIdx.x;
    int b = blk >> 3, sp = blk & 7;
    int wid = threadIdx.x >> 5, lane = threadIdx.x & 31;
    int m = lane & 15;
    int kb = (lane >> 4) * 2;
    const float* arow = cpm + ((size_t)b * LSEQ + sp * 16 + m) * H;
    const float* brow = chm + ((size_t)b * LSEQ + wid * 16 + m) * H;
    v8f acc = {};
    for (int k0 = 0; k0 < H; k0 += 4) {
        v2f a, bb;
        a.x = arow[k0 + kb];  a.y = arow[k0 + kb + 1];
        bb.x = brow[k0 + kb]; bb.y = brow[k0 + kb + 1];
        acc = __builtin_amdgcn_wmma_f32_16x16x4_f32(false, a, false, bb, (short)0, acc, false, false);
    }
    int hi = lane >> 4, n = lane & 15;
#pragma unroll
    for (int r = 0; r < 8; r++) {
        int pm = sp * 16 + r + hi * 8;
        int qn = wid * 16 + n;
        float denom = fmaxf(np_[b * LSEQ + pm] * nh_[b * LSEQ + qn], EPS);
        cosm[((size_t)b * LSEQ + pm) * LSEQ + qn] = acc[r] / denom;
    }
}

// ---------------- K3: cos masked max/mean + row/col sums -----------------
// grid = B*LSEQ*2, block = 128
__global__ void k_cosred(const float* __restrict__ cosm,
                         const float* __restrict__ mask_p, const float* __restrict__ mask_h,
                         const float* __restrict__ summ_p, const float* __restrict__ summ_h,
                         float* __restrict__ rowsum, float* __restrict__ colsum,
                         float* __restrict__ out) {
    int i = blockIdx.x;
    int side = i & 1;        // 0: p rows (reduce over q), 1: h cols (reduce over p)
    int bl = i >> 1;
    int b = bl / LSEQ, l = bl % LSEQ;
    int t = threadIdx.x;
    float v, om;
    if (side == 0) { v = cosm[((size_t)b * LSEQ + l) * LSEQ + t]; om = mask_h[b * LSEQ + t]; }
    else           { v = cosm[((size_t)b * LSEQ + t) * LSEQ + l]; om = mask_p[b * LSEQ + t]; }
    __shared__ float smax[128], ssum[128];
    smax[t] = (om > 0.f) ? v : NEGV;
    ssum[t] = v;
    __syncthreads();
    for (int o = 64; o > 0; o >>= 1) {
        if (t < o) { smax[t] = fmaxf(smax[t], smax[t + o]); ssum[t] += ssum[t + o]; }
        __syncthreads();
    }
    if (t == 0) {
        float myMask = (side == 0) ? mask_p[bl] : mask_h[bl];
        float osum   = (side == 0) ? summ_h[b] : summ_p[b];
        float* orow = out + ((size_t)(side ? B * LSEQ : 0) + bl) * DOUT;
        orow[0] = smax[0] * myMask;
        orow[1] = ssum[0] / fmaxf(myMask * osum, EPS);
        if (side == 0) rowsum[bl] = ssum[0]; else colsum[bl] = ssum[0];
    }
}

// ---------------- K4: attentive means via WMMA -----------------
// grid = B*2*13, block = 256 (8 waves -> 8 row strips), 13 col tiles over H=200
__global__ void k_attmean(const float* __restrict__ cosm,
                          const float* __restrict__ cpm, const float* __restrict__ chm,
                          const float* __restrict__ rowsum, const float* __restrict__ colsum,
                          float* __restrict__ amh, float* __restrict__ amp) {
    int i = blockIdx.x;
    int ct = i % 13;
    int side = (i / 13) & 1;
    int b = i / 26;
    int wid = threadIdx.x >> 5, lane = threadIdx.x & 31;
    int m = lane & 15, kb = (lane >> 4) * 2, n = lane & 15, hi = lane >> 4;
    int mrow = wid * 16 + m;
    int ncol = ct * 16 + n;
    const float* cb = cosm + (size_t)b * LSEQ * LSEQ;
    const float* xb = ((side == 0) ? chm : cpm) + (size_t)b * LSEQ * H;
    v8f acc = {};
    for (int k0 = 0; k0 < LSEQ; k0 += 4) {
        v2f a, bb;
        if (side == 0) {
            a.x = cb[(size_t)mrow * LSEQ + k0 + kb];
            a.y = cb[(size_t)mrow * LSEQ + k0 + kb + 1];
        } else {
            a.x = cb[(size_t)(k0 + kb) * LSEQ + mrow];
            a.y = cb[(size_t)(k0 + kb + 1) * LSEQ + mrow];
        }
        bb.x = (ncol < H) ? xb[(size_t)(k0 + kb) * H + ncol] : 0.f;
        bb.y = (ncol < H) ? xb[(size_t)(k0 + kb + 1) * H + ncol] : 0.f;
        acc = __builtin_amdgcn_wmma_f32_16x16x4_f32(false, a, false, bb, (short)0, acc, false, false);
    }
    if (ncol < H) {
        const float* dsum = (side == 0) ? rowsum : colsum;
        float* outp = (side == 0) ? amh : amp;
#pragma unroll
        for (int r = 0; r < 8; r++) {
            int row = wid * 16 + r + hi * 8;
            float dv = fmaxf(dsum[b * LSEQ + row], EPS);
            outp[((size_t)b * LSEQ + row) * H + ncol] = acc[r] / dv;
        }
    }
}

// ---------------- K5: attentive max vectors -----------------
// grid = B*LSEQ*2, block = 256
__global__ void k_attmax(const float* __restrict__ cosm,
                         const float* __restrict__ cpm, const float* __restrict__ chm,
                         const float* __restrict__ mask_p, const float* __restrict__ mask_h,
                         float* __restrict__ axh, float* __restrict__ axp) {
    int i = blockIdx.x;
    int side = i & 1; int bl = i >> 1;
    int b = bl / LSEQ, l = bl % LSEQ;
    __shared__ float crow[LSEQ];
    __shared__ float msk[LSEQ];
    int t = threadIdx.x;
    if (t < LSEQ) {
        if (side == 0) { crow[t] = cosm[((size_t)b * LSEQ + l) * LSEQ + t]; msk[t] = mask_h[b * LSEQ + t]; }
        else           { crow[t] = cosm[((size_t)b * LSEQ + t) * LSEQ + l]; msk[t] = mask_p[b * LSEQ + t]; }
    }
    __syncthreads();
    float myMask = (side == 0) ? mask_p[bl] : mask_h[bl];
    const float* xb = ((side == 0) ? chm : cpm) + (size_t)b * LSEQ * H;
    float* outp = ((side == 0) ? axh : axp) + (size_t)bl * H;
    for (int h = t; h < H; h += blockDim.x) {
        float mx = NEGV;
        for (int q = 0; q < LSEQ; q++) {
            if (msk[q] > 0.f) mx = fmaxf(mx, xb[(size_t)q * H + h] * crow[q]);
        }
        outp[h] = mx * myMask;
    }
}

// ---------------- K6: multi-perspective match (full / att / max-att) -----------------
// grid = B*LSEQ*2*3, block = 32
__global__ void k_mpm(const float* __restrict__ cpm, const float* __restrict__ chm,
                      const float* __restrict__ last_p, const float* __restrict__ last_h,
                      const float* __restrict__ amh, const float* __restrict__ amp,
                      const float* __restrict__ axh, const float* __restrict__ axp,
                      const float* __restrict__ w_full, const float* __restrict__ w_att,
                      const float* __restrict__ w_max_att,
                      float* __restrict__ out) {
    int i = blockIdx.x;
    int type = i % 3;
    int side = (i / 3) & 1;
    int bl = i / 6;
    int b = bl / LSEQ;
    int lane = threadIdx.x;
    const float* v1 = ((side == 0) ? cpm : chm) + (size_t)bl * H;
    const float* v2; const float* w;
    if (type == 0)      { v2 = ((side == 0) ? last_h : last_p) + (size_t)b * H; w = w_full; }
    else if (type == 1) { v2 = ((side == 0) ? amh : amp) + (size_t)bl * H;      w = w_att; }
    else                { v2 = ((side == 0) ? axh : axp) + (size_t)bl * H;      w = w_max_att; }

    float d = 0.f, s1 = 0.f, s2 = 0.f;
    float wd[NP], wa1[NP], wa2[NP];
#pragma unroll
    for (int k = 0; k < NP; k++) { wd[k] = 0.f; wa1[k] = 0.f; wa2[k] = 0.f; }
    for (int h = lane; h < H; h += 32) {
        float x = v1[h], y = v2[h];
        d += x * y; s1 += x * x; s2 += y * y;
#pragma unroll
        for (int k = 0; k < NP; k++) {
            float wk = w[k * H + h]; float ww = wk * wk;
            wd[k] += ww * x * y; wa1[k] += ww * x * x; wa2[k] += ww * y * y;
        }
    }
    d = wsum32(d); s1 = wsum32(s1); s2 = wsum32(s2);
#pragma unroll
    for (int k = 0; k < NP; k++) { wd[k] = wsum32(wd[k]); wa1[k] = wsum32(wa1[k]); wa2[k] = wsum32(wa2[k]); }
    if (lane == 0) {
        float* orow = out + ((size_t)(side ? B * LSEQ : 0) + bl) * DOUT;
        const int m1c = (type == 0) ? 2 : ((type == 1) ? 63 : 84);
        const int mmc = (type == 0) ? 3 : ((type == 1) ? 64 : 85);
        orow[m1c] = d / fmaxf(sqrtf(s1) * sqrtf(s2), EPS);
#pragma unroll
        for (int k = 0; k < NP; k++)
            orow[mmc + k] = wd[k] / fmaxf(sqrtf(wa1[k]) * sqrtf(wa2[k]), EPS);
    }
}

// ---------------- K7: maxpool match via weighted WMMA GEMM + fused reduce -----------------
// grid = B*NP, block = 256 (8 waves -> 8 row tiles, each loops 8 col tiles)
__global__ void k_maxpool(const float* __restrict__ cpm, const float* __restrict__ chm,
                          const float* __restrict__ wmp,
                          const float* __restrict__ wnp, const float* __restrict__ wnh,
                          const float* __restrict__ mask_p, const float* __restrict__ mask_h,
                          const float* __restrict__ summ_p, const float* __restrict__ summ_h,
                          float* __restrict__ out) {
    int b = blockIdx.x / NP;
    int kidx = blockIdx.x % NP;
    __shared__ float wl[H];
    __shared__ unsigned rmaxU[LSEQ]; __shared__ float rsum[LSEQ];
    __shared__ unsigned cmaxU[LSEQ]; __shared__ float csum[LSEQ];
    int t = threadIdx.x;
    for (int j = t; j < H; j += 256) wl[j] = wmp[kidx * H + j];
    if (t < LSEQ) { rmaxU[t] = fkey(NEGV); rsum[t] = 0.f; cmaxU[t] = fkey(NEGV); csum[t] = 0.f; }
    __syncthreads();
    int wid = t >> 5, lane = t & 31;
    int m = lane & 15, kb = (lane >> 4) * 2, n = lane & 15, hi = lane >> 4;
    int tp = wid;
    const float* arow = cpm + ((size_t)b * LSEQ + tp * 16 + m) * H;
    const float* wnprow = wnp + ((size_t)b * NP + kidx) * LSEQ;
    const float* wnhrow = wnh + ((size_t)b * NP + kidx) * LSEQ;
    for (int tq = 0; tq < 8; tq++) {
        const float* brow = chm + ((size_t)b * LSEQ + tq * 16 + n) * H;
        v8f acc = {};
        for (int k0 = 0; k0 < H; k0 += 4) {
            v2f a, bb;
            float w0 = wl[k0 + kb], w1 = wl[k0 + kb + 1];
            a.x = arow[k0 + kb] * w0;  a.y = arow[k0 + kb + 1] * w1;
            bb.x = brow[k0 + kb] * w0; bb.y = brow[k0 + kb + 1] * w1;
            acc = __builtin_amdgcn_wmma_f32_16x16x4_f32(false, a, false, bb, (short)0, acc, false, false);
        }
        int q = tq * 16 + n;
        float mh = mask_h[b * LSEQ + q];
        float cAcc = 0.f, cMax = NEGV;
#pragma unroll
        for (int r = 0; r < 8; r++) {
            int p = tp * 16 + r + hi * 8;
            float mp = mask_p[b * LSEQ + p];
            float val = acc[r] / fmaxf(wnprow[p] * wnhrow[q], EPS);
            bool vm = (mp > 0.f) && (mh > 0.f);
            float vs = vm ? val : 0.f;
            float vx = vm ? val : NEGV;
            atomicAdd(&rsum[p], vs);
            atomicMax(&rmaxU[p], fkey(vx));
            cAcc += vs; cMax = fmaxf(cMax, vx);
        }
        atomicAdd(&csum[q], cAcc);
        atomicMax(&cmaxU[q], fkey(cMax));
    }
    __syncthreads();
    if (t < LSEQ) {
        int p = t;
        float mp = mask_p[b * LSEQ + p];
        float* orow = out + ((size_t)b * LSEQ + p) * DOUT;
        orow[23 + kidx] = funkey(rmaxU[p]) * mp;
        orow[43 + kidx] = rsum[p] / fmaxf(mp * summ_h[b], EPS);
    } else {
        int q = t - LSEQ;
        float mh = mask_h[b * LSEQ + q];
        float* orow = out + (size_t)B * LSEQ * DOUT + ((size_t)b * LSEQ + q) * DOUT;
        orow[23 + kidx] = funkey(cmaxU[q]) * mh;
        orow[43 + kidx] = csum[q] / fmaxf(mh * summ_p[b], EPS);
    }
}

extern "C" void kernel_launch(void* const* d_in, const int* in_sizes, int n_in,
                              void* d_out, int out_size, void* d_ws, size_t ws_size,
                              hipStream_t stream) {
    (void)in_sizes; (void)n_in; (void)out_size; (void)ws_size;
    const float* context_p = (const float*)d_in[0];
    const float* mask_p    = (const float*)d_in[1];
    const float* context_h = (const float*)d_in[2];
    const float* mask_h    = (const float*)d_in[3];
    const float* w_full    = (const float*)d_in[4];
    const float* w_maxpool = (const float*)d_in[5];
    const float* w_att     = (const float*)d_in[6];
    const float* w_max_att = (const float*)d_in[7];
    float* out = (float*)d_out;

    float* ws = (float*)d_ws;
    const size_t BLH = (size_t)B * LSEQ * H;      // 819200
    const size_t BLL = (size_t)B * LSEQ * LSEQ;   // 524288
    const size_t BL  = (size_t)B * LSEQ;          // 4096
    float* cp_m   = ws;                  ws += BLH;
    float* ch_m   = ws;                  ws += BLH;
    float* cosm   = ws;                  ws += BLL;
    float* n_p    = ws;                  ws += BL;
    float* n_h    = ws;                  ws += BL;
    float* wn_p   = ws;                  ws += (size_t)B * NP * LSEQ;
    float* wn_h   = ws;                  ws += (size_t)B * NP * LSEQ;
    float* rowsum = ws;                  ws += BL;
    float* colsum = ws;                  ws += BL;
    float* last_p = ws;                  ws += (size_t)B * H;
    float* last_h = ws;                  ws += (size_t)B * H;
    float* amh    = ws;                  ws += BLH;
    float* amp    = ws;                  ws += BLH;
    float* axh    = ws;                  ws += BLH;
    float* axp    = ws;                  ws += BLH;
    float* summ_p = ws;                  ws += B;
    float* summ_h = ws;                  ws += B;

    k_pre<<<B * LSEQ, 32, 0, stream>>>(context_p, mask_p, w_maxpool, cp_m, n_p, wn_p);
    k_pre<<<B * LSEQ, 32, 0, stream>>>(context_h, mask_h, w_maxpool, ch_m, n_h, wn_h);
    k_last<<<B, 256, 0, stream>>>(mask_p, mask_h, cp_m, ch_m, last_p, last_h, summ_p, summ_h);
    k_cos<<<B * 8, 256, 0, stream>>>(cp_m, ch_m, n_p, n_h, cosm);
    k_cosred<<<B * LSEQ * 2, 128, 0, stream>>>(cosm, mask_p, mask_h, summ_p, summ_h,
                                               rowsum, colsum, out);
    k_attmean<<<B * 2 * 13, 256, 0, stream>>>(cosm, cp_m, ch_m, rowsum, colsum, amh, amp);
    k_attmax<<<B * LSEQ * 2, 256, 0, stream>>>(cosm, cp_m, ch_m, mask_p, mask_h, axh, axp);
    k_mpm<<<B * LSEQ * 2 * 3, 32, 0, stream>>>(cp_m, ch_m, last_p, last_h, amh, amp, axh, axp,
                                               w_full, w_att, w_max_att, out);
    k_maxpool<<<B * NP, 256, 0, stream>>>(cp_m, ch_m, w_maxpool, wn_p, wn_h,
                                          mask_p, mask_h, summ_p, summ_h, out);
}